// MultiHeadGATLayer_3435973837173
// MI455X (gfx1250) — compile-verified
//
#include <hip/hip_runtime.h>
#include <hip/hip_bf16.h>

typedef __attribute__((ext_vector_type(2))) float v2f;
typedef __attribute__((ext_vector_type(8))) float v8f;

#define NEG_SLOPE 0.2f
#define CTOT 256   // H*Cout
#define CIN  128

// ---------- helpers ----------
__device__ __forceinline__ void atomicMaxF32(float* addr, float val) {
    // monotonic float-bits trick; init value is -inf so every path is safe
    if (val >= 0.0f)
        atomicMax(reinterpret_cast<int*>(addr), __float_as_int(val));
    else
        atomicMin(reinterpret_cast<unsigned int*>(addr),
                  static_cast<unsigned int>(__float_as_int(val)));
}

// ---------- 1. repack W (H,Cout,Cin) -> Wt (Cin, H*Cout) ----------
__global__ void repack_w_kernel(const float* __restrict__ W, float* __restrict__ Wt) {
    int idx = blockIdx.x * blockDim.x + threadIdx.x;       // over CIN*CTOT
    if (idx >= CIN * CTOT) return;
    int k = idx / CTOT;
    int c = idx - k * CTOT;                                // c = h*64 + o
    Wt[idx] = W[(size_t)c * CIN + k];
}

// ---------- 2. init m/-inf, denom/0 ----------
__global__ void init_kernel(float* __restrict__ m, float* __restrict__ denom, int n) {
    int i = blockIdx.x * blockDim.x + threadIdx.x;
    if (i < n) { m[i] = -__builtin_inff(); denom[i] = 0.0f; }
}

// ---------- 3. xh = x @ Wt via V_WMMA_F32_16X16X4_F32 ----------
__global__ __launch_bounds__(32) void gemm_xh_kernel(const float* __restrict__ x,
                                                     const float* __restrict__ Wt,
                                                     float* __restrict__ xh, int N) {
    const int lane = threadIdx.x;          // wave32
    const int m0   = blockIdx.x * 16;
    const int n0   = blockIdx.y * 16;
    const int ml   = lane & 15;            // M (A) / N (B,C,D) lane index
    const int half = lane >> 4;

    int arow = m0 + ml; if (arow >= N) arow = N - 1;       // clamp (N%16==0 normally)
    const float* xrow = x + (size_t)arow * CIN;

    v8f c = {};
    #pragma unroll
    for (int k = 0; k < CIN; k += 4) {
        // A 16x4 f32: VGPR v, half h -> K = v + 2*h, M = lane&15
        v2f a;
        a.x = xrow[k + 2 * half + 0];
        a.y = xrow[k + 2 * half + 1];
        // B 4x16 f32: VGPR v, half h -> K = v + 2*h, N = lane&15
        v2f b;
        b.x = Wt[(size_t)(k + 2 * half + 0) * CTOT + n0 + ml];
        b.y = Wt[(size_t)(k + 2 * half + 1) * CTOT + n0 + ml];
        c = __builtin_amdgcn_wmma_f32_16x16x4_f32(false, a, false, b,
                                                  (short)0, c, false, false);
    }
    // D 16x16 f32: VGPR v -> M = v + 8*half, N = lane&15
    #pragma unroll
    for (int v = 0; v < 8; ++v) {
        int row = m0 + v + 8 * half;
        if (row < N) xh[(size_t)row * CTOT + n0 + ml] = c[v];
    }
}

// ---------- 4. attention scores per (node, head) ----------
__global__ void scores_kernel(const float* __restrict__ xh, const float* __restrict__ attn,
                              float* __restrict__ s_dst, float* __restrict__ s_src, int NH) {
    int idx = blockIdx.x * blockDim.x + threadIdx.x;       // over N*4
    if (idx >= NH) return;
    int n = idx >> 2, h = idx & 3;
    const float* row = xh + (size_t)n * CTOT + h * 64;
    const float* ai  = attn + h * 128;                     // a[:, :Cout]
    const float* aj  = ai + 64;                            // a[:, Cout:]
    float sd = 0.0f, ss = 0.0f;
    #pragma unroll
    for (int o = 0; o < 64; ++o) {
        float v = row[o];
        sd = __fmaf_rn(v, ai[o], sd);
        ss = __fmaf_rn(v, aj[o], ss);
    }
    s_dst[idx] = sd;
    s_src[idx] = ss;
}

// ---------- 5. per-edge leaky score -> segment max ----------
__global__ void edge_max_kernel(const int* __restrict__ ei,
                                const float* __restrict__ s_dst,
                                const float* __restrict__ s_src,
                                float* __restrict__ m, int E, int Etot) {
    int e = blockIdx.x * blockDim.x + threadIdx.x;
    if (e >= Etot) return;
    int r, c;
    if (e < E) { r = ei[e]; c = ei[E + e]; } else { r = c = e - E; }
    #pragma unroll
    for (int h = 0; h < 4; ++h) {
        float t = s_dst[(size_t)c * 4 + h] + s_src[(size_t)r * 4 + h];
        t = t > 0.0f ? t : NEG_SLOPE * t;
        atomicMaxF32(&m[(size_t)c * 4 + h], t);
    }
}

// ---------- 6. per-edge exp(e - m) -> segment sum ----------
__global__ void edge_denom_kernel(const int* __restrict__ ei,
                                  const float* __restrict__ s_dst,
                                  const float* __restrict__ s_src,
                                  const float* __restrict__ m,
                                  float* __restrict__ denom, int E, int Etot) {
    int e = blockIdx.x * blockDim.x + threadIdx.x;
    if (e >= Etot) return;
    int r, c;
    if (e < E) { r = ei[e]; c = ei[E + e]; } else { r = c = e - E; }
    #pragma unroll
    for (int h = 0; h < 4; ++h) {
        float t = s_dst[(size_t)c * 4 + h] + s_src[(size_t)r * 4 + h];
        t = t > 0.0f ? t : NEG_SLOPE * t;
        float ex = __expf(t - m[(size_t)c * 4 + h]);
        unsafeAtomicAdd(&denom[(size_t)c * 4 + h], ex);
    }
}

// ---------- 7. wave32-per-edge aggregation: out[col] += alpha * xh[row] ----------
__global__ __launch_bounds__(256) void aggregate_kernel(const int* __restrict__ ei,
                                                        const float* __restrict__ xh,
                                                        const float* __restrict__ s_dst,
                                                        const float* __restrict__ s_src,
                                                        const float* __restrict__ m,
                                                        const float* __restrict__ denom,
                                                        float* __restrict__ out,
                                                        int E, int Etot) {
    int wave = (int)((blockIdx.x * (unsigned)blockDim.x + threadIdx.x) >> 5);
    int lane = threadIdx.x & 31;
    if (wave >= Etot) return;
    int r, c;
    if (wave < E) { r = ei[wave]; c = ei[E + wave]; } else { r = c = wave - E; }

    int c0 = lane * 8;           // 8 contiguous cols per lane -> single head per lane
    int h  = c0 >> 6;

    float t = s_dst[(size_t)c * 4 + h] + s_src[(size_t)r * 4 + h];
    t = t > 0.0f ? t : NEG_SLOPE * t;
    float alpha = __expf(t - m[(size_t)c * 4 + h]) / denom[(size_t)c * 4 + h];

    const float4* src = reinterpret_cast<const float4*>(xh + (size_t)r * CTOT + c0);
    float4 v0 = src[0];
    float4 v1 = src[1];
    float* dst = out + (size_t)c * CTOT + c0;
    unsafeAtomicAdd(dst + 0, alpha * v0.x);
    unsafeAtomicAdd(dst + 1, alpha * v0.y);
    unsafeAtomicAdd(dst + 2, alpha * v0.z);
    unsafeAtomicAdd(dst + 3, alpha * v0.w);
    unsafeAtomicAdd(dst + 4, alpha * v1.x);
    unsafeAtomicAdd(dst + 5, alpha * v1.y);
    unsafeAtomicAdd(dst + 6, alpha * v1.z);
    unsafeAtomicAdd(dst + 7, alpha * v1.w);
}

extern "C" void kernel_launch(void* const* d_in, const int* in_sizes, int n_in,
                              void* d_out, int out_size, void* d_ws, size_t ws_size,
                              hipStream_t stream) {
    const float* x    = (const float*)d_in[0];
    const int*   ei   = (const int*)d_in[1];
    const float* W    = (const float*)d_in[2];
    const float* attn = (const float*)d_in[3];

    const int N    = out_size / CTOT;       // 50000
    const int E    = in_sizes[1] / 2;       // 800000
    const int Etot = E + N;                 // + self loops
    float* outp = (float*)d_out;

    // workspace layout (floats)
    float* ws    = (float*)d_ws;
    float* xh    = ws;                                  // N*256
    float* Wt    = xh + (size_t)N * CTOT;               // 128*256
    float* s_dst = Wt + (size_t)CIN * CTOT;             // N*4
    float* s_src = s_dst + (size_t)N * 4;               // N*4
    float* mbuf  = s_src + (size_t)N * 4;               // N*4
    float* denom = mbuf + (size_t)N * 4;                // N*4

    hipMemsetAsync(d_out, 0, (size_t)out_size * sizeof(float), stream);

    repack_w_kernel<<<(CIN * CTOT + 255) / 256, 256, 0, stream>>>(W, Wt);
    init_kernel<<<(N * 4 + 255) / 256, 256, 0, stream>>>(mbuf, denom, N * 4);

    dim3 ggrid((N + 15) / 16, CTOT / 16);
    gemm_xh_kernel<<<ggrid, 32, 0, stream>>>(x, Wt, xh, N);

    scores_kernel<<<(N * 4 + 255) / 256, 256, 0, stream>>>(xh, attn, s_dst, s_src, N * 4);

    edge_max_kernel<<<(Etot + 255) / 256, 256, 0, stream>>>(ei, s_dst, s_src, mbuf, E, Etot);
    edge_denom_kernel<<<(Etot + 255) / 256, 256, 0, stream>>>(ei, s_dst, s_src, mbuf, denom, E, Etot);

    aggregate_kernel<<<(Etot + 7) / 8, 256, 0, stream>>>(ei, xh, s_dst, s_src,
                                                         mbuf, denom, outp, E, Etot);
}